// HebbianNet_34780645163328
// MI455X (gfx1250) — compile-verified
//
#include <hip/hip_runtime.h>
#include <hip/hip_bf16.h>

typedef float v2f __attribute__((ext_vector_type(2)));
typedef float v8f __attribute__((ext_vector_type(8)));

#define HEBB_RATE 1e-3f

// ---------------------------------------------------------------------------
// Kernel 1: vj[b,o] = relu( sum_k vi[b,k] * W[o,k] + bias[o] )
// One wave (32 threads) per 16-wide output-column tile.
// Uses V_WMMA_F32_16X16X4_F32 (exact fp32, matches reference dtype).
//   A fragment (16x4):  lane L -> M = L%16, VGPR j holds K = (L/16)*2 + j
//   B fragment (4x16):  lane L -> N = L%16, VGPR j holds K = (L/16)*2 + j
//   C/D (16x16):        VGPR r, lanes 0-15 -> (M=r, N=lane),
//                               lanes 16-31 -> (M=r+8, N=lane-16)
//
// Batch is padded M=Brows..15. Padding lanes read a CLAMPED valid row (no
// OOB, no exec-mask branches) and are NOT zeroed: in D = A*B + C, D row M
// depends only on A row M, so garbage in pad rows only reaches D rows
// 8..15, which are never stored. WMMA raises no FP exceptions (ISA §7.12),
// so the duplicated-row values are harmless. This keeps the inner loop to
// pure { global_load_b64 x2 ; v_wmma } with EXEC all-1s throughout.
// ---------------------------------------------------------------------------
__global__ __launch_bounds__(32) void wmma_gemm_relu(
    const float* __restrict__ vi,   // [Brows, K]
    const float* __restrict__ W,    // [Out, K]   (row-major, we use W^T)
    const float* __restrict__ bias, // [Out]
    float* __restrict__ vj,         // [Brows, Out]
    int Brows, int K, int Out)
{
  const int lane  = threadIdx.x;   // 0..31
  const int m     = lane & 15;     // A row / B column within tile
  const int khalf = lane >> 4;     // 0 or 1  -> K pair {0,1} vs {2,3}
  const int n0    = blockIdx.x * 16;
  const int ncol  = n0 + m;

  const int mrow = (m < Brows) ? m : (Brows - 1);   // clamp pad rows in-bounds

  const float* aRow = vi + (size_t)mrow * K + 2 * khalf;
  const float* bRow = W  + (size_t)ncol * K + 2 * khalf;

  v8f c = {};
  for (int k = 0; k < K; k += 4) {
    v2f a = *(const v2f*)(aRow + k);   // global_load_b64
    v2f b = *(const v2f*)(bRow + k);   // global_load_b64
    // (neg_a, A, neg_b, B, c_mod, C, reuse_a, reuse_b)
    c = __builtin_amdgcn_wmma_f32_16x16x4_f32(
        false, a, false, b, (short)0, c, false, false);
  }

  if (khalf == 0) {                 // lanes 0..15 hold rows M = 0..7
    const float bo = bias[ncol];
#pragma unroll
    for (int r = 0; r < 8; ++r) {
      if (r < Brows) {
        float v = c[r] + bo;
        vj[(size_t)r * Out + ncol] = v > 0.0f ? v : 0.0f;
      }
    }
  }
}

// ---------------------------------------------------------------------------
// Kernel 2: Hebbian per-edge update (irreducibly elementwise; VALU path —
// the ReLU inside the channel sum blocks any matmul factorization).
//   out[b,o] = vj[b,o] + scale * ( sum_i vi[b,i]*dot_i  +  bc2 * sum_i vi[b,i] )
//   dot_i    = sum_h Wc2[h] * relu( Wc1[h,0]*vi[b,i] + Wc1[h,1]*W[o,i]
//                                   + Wc1[h,2]*vj[b,o] + bc1[h] )
// One wave32 per (b,o); lanes stride over i; butterfly reduction at the end.
// The (c_h*vj+bc1_h) term is constant per (b,o) and hoisted out of the loop.
// ---------------------------------------------------------------------------
__global__ __launch_bounds__(256) void hebb_update(
    const float* __restrict__ vi,   // [B, In]
    const float* __restrict__ W,    // [Out, In]
    const float* __restrict__ vj,   // [B, Out]
    const float* __restrict__ Wc1,  // [8, 3]
    const float* __restrict__ bc1,  // [8]
    const float* __restrict__ Wc2,  // [8]   (row 0 of [1,8])
    const float* __restrict__ bc2,  // [1]
    const int*   __restrict__ batch_num, // [1]
    float* __restrict__ out,        // [B, Out]
    int B, int In, int Out)
{
  const int lane = threadIdx.x & 31;
  const int wid  = (int)((blockIdx.x * blockDim.x + threadIdx.x) >> 5);
  const int total = B * Out;
  if (wid >= total) return;

  const int b = wid / Out;
  const int o = wid - b * Out;

  const float scale = HEBB_RATE / (float)(*batch_num);
  const float vjo   = vj[(size_t)b * Out + o];

  float ca[8], cw[8], cg[8], cq[8];
#pragma unroll
  for (int h = 0; h < 8; ++h) {
    ca[h] = Wc1[h * 3 + 0];
    cw[h] = Wc1[h * 3 + 1];
    cg[h] = fmaf(Wc1[h * 3 + 2], vjo, bc1[h]);  // gamma_h (const per (b,o))
    cq[h] = Wc2[h];
  }

  const float* Wrow  = W  + (size_t)o * In;
  const float* virow = vi + (size_t)b * In;

  float acc = 0.0f;   // sum_i vi * dot_i
  float sv  = 0.0f;   // sum_i vi          (for the bc2 term)
  for (int i = lane; i < In; i += 32) {
    const float v  = virow[i];
    const float ww = Wrow[i];
    float s = 0.0f;
#pragma unroll
    for (int h = 0; h < 8; ++h) {
      float t = fmaf(ca[h], v, fmaf(cw[h], ww, cg[h]));
      t = t > 0.0f ? t : 0.0f;
      s = fmaf(cq[h], t, s);
    }
    acc = fmaf(v, s, acc);
    sv += v;
  }

  // wave32 butterfly reduction
#pragma unroll
  for (int off = 16; off > 0; off >>= 1) {
    acc += __shfl_xor(acc, off, 32);
    sv  += __shfl_xor(sv,  off, 32);
  }

  if (lane == 0) {
    const float delta = scale * fmaf(bc2[0], sv, acc);
    out[(size_t)b * Out + o] = vjo + delta;
  }
}

// ---------------------------------------------------------------------------
// Launch: two layer_step invocations.
//   layer1: vi = x    [8,512],  W1 [1024,512] -> out1 [8,1024]
//   layer2: vi = out1 [8,1024], W2 [512,1024] -> out  [8,512]
// Workspace layout (floats): vj1[8*1024] | out1[8*1024] | vj2[8*512]
// ---------------------------------------------------------------------------
extern "C" void kernel_launch(void* const* d_in, const int* in_sizes, int n_in,
                              void* d_out, int out_size, void* d_ws, size_t ws_size,
                              hipStream_t stream) {
  (void)in_sizes; (void)n_in; (void)out_size; (void)ws_size;

  const float* x   = (const float*)d_in[0];
  const float* W1  = (const float*)d_in[1];
  const float* b1  = (const float*)d_in[2];
  const float* W2  = (const float*)d_in[3];
  const float* b2  = (const float*)d_in[4];
  const float* Wc1 = (const float*)d_in[5];
  const float* bc1 = (const float*)d_in[6];
  const float* Wc2 = (const float*)d_in[7];
  const float* bc2 = (const float*)d_in[8];
  const int*   bn  = (const int*)d_in[9];

  const int B = 8, I = 512, H = 1024, O = 512;

  float* ws   = (float*)d_ws;
  float* vj1  = ws;               // [8,1024]
  float* out1 = vj1 + B * H;      // [8,1024]
  float* vj2  = out1 + B * H;     // [8,512]
  float* outf = (float*)d_out;    // [8,512]

  // ---- layer 1 ----
  wmma_gemm_relu<<<H / 16, 32, 0, stream>>>(x, W1, b1, vj1, B, I, H);
  {
    const int waves = B * H;                 // one wave32 per (b,o)
    const int blocks = (waves + 7) / 8;      // 8 waves per 256-thread block
    hebb_update<<<blocks, 256, 0, stream>>>(x, W1, vj1, Wc1, bc1, Wc2, bc2, bn,
                                            out1, B, I, H);
  }

  // ---- layer 2 ----
  wmma_gemm_relu<<<O / 16, 32, 0, stream>>>(out1, W2, b2, vj2, B, H, O);
  {
    const int waves = B * O;
    const int blocks = (waves + 7) / 8;
    hebb_update<<<blocks, 256, 0, stream>>>(out1, W2, vj2, Wc1, bc1, Wc2, bc2, bn,
                                            outf, B, H, O);
  }
}